// CTAModule_50242527429181
// MI455X (gfx1250) — compile-verified
//
#include <hip/hip_runtime.h>
#include <stdint.h>

typedef __attribute__((ext_vector_type(2))) float    v2f;
typedef __attribute__((ext_vector_type(4))) float    v4f;
typedef __attribute__((ext_vector_type(8))) float    v8f;
typedef __attribute__((ext_vector_type(4))) unsigned u32x4;
typedef __attribute__((ext_vector_type(8))) int      i32x8;
typedef __attribute__((ext_vector_type(4))) int      i32x4;

#define T     16       // time steps
#define CCH   64       // channels
#define HW    3136     // 56*56
#define WIMG  56
#define P     64       // DIM*DIM token features
#define QKD   128      // 2*DIM_HEAD
#define CHW   (CCH*HW) // t-stride in x (elements)
#define SLABSTRIDE 76  // 64 data + 12 pad dwords per row: conflict-free for rows +2 and +8

#if defined(__gfx1250__) && __has_builtin(__builtin_amdgcn_tensor_load_to_lds)
#define USE_TDM 1
#else
#define USE_TDM 0
#endif

__global__ __launch_bounds__(256) void cta_attn_kernel(
    const float* __restrict__ x,
    const float* __restrict__ pos,
    const float* __restrict__ Wqk,
    const float* __restrict__ gamma,
    const float* __restrict__ beta,
    float* __restrict__ out)
{
  // colsum (phase A pooling) and slabs (phase B v-staging) have disjoint lifetimes.
  __shared__ __align__(16) union {
    float colsum[T * 8 * WIMG];        // 7168 floats: 7-row column sums
    float slab[8 * 16 * SLABSTRIDE];   // 9728 floats: per-wave 16x64 v tile (stride 76)
  } u;
  __shared__ float s_lds[T * 65];      // tokens (padded stride)
  __shared__ float qk_lds[T * 129];    // q|k (padded stride)
  __shared__ float attn_lds[T * 17];   // attention matrix (padded stride)
  __shared__ float mu_lds[T];
  __shared__ float rs_lds[T];

  const int tid    = threadIdx.x;
  const int lane   = tid & 31;
  const int wave   = tid >> 5;
  const int lane16 = lane & 15;
  const int half   = lane >> 4;

  const int bc = blockIdx.x;           // b*64 + c
  const int bb = bc >> 6;
  const int ch = bc & 63;
  const size_t xbase = ((size_t)(bb * T) * CCH + ch) * HW;  // x[b,0,c,0,0]

  // ---------- Phase A1: 7-row column sums, float4 (b128) coalesced loads
  // 1792 float4 column groups: (t, py, c4) with c4 in [0,14)
#pragma unroll
  for (int i = 0; i < 7; ++i) {
    int g  = tid + 256 * i;            // 0..1791
    int c4 = g % 14;
    int py = (g / 14) & 7;
    int t  = g / 112;
    const float* src = x + xbase + (size_t)t * CHW + (py * 7) * WIMG + c4 * 4;
    v4f s = {};
#pragma unroll
    for (int r = 0; r < 7; ++r)
      s += *(const v4f*)(src + r * WIMG);
    *(v4f*)(u.colsum + (t * 8 + py) * WIMG + c4 * 4) = s;
  }
  __syncthreads();

  // ---------- Phase A2: horizontal 7-sum + pos embedding -> tokens
  for (int i = 0; i < 4; ++i) {
    int o = tid + 256 * i;
    int t = o >> 6, p = o & 63;
    int py = p >> 3, px = p & 7;
    const float* cs = u.colsum + (t * 8 + py) * WIMG + px * 7;
    float s = 0.f;
#pragma unroll
    for (int cc = 0; cc < 7; ++cc) s += cs[cc];
    s_lds[t * 65 + p] = s * (1.0f / 49.0f) + pos[((size_t)bc * T + t) * P + p];
  }
  __syncthreads();

  // ---------- Phase A3: LayerNorm over 64 features
  if (tid < T) {
    float m = 0.f;
    for (int p = 0; p < P; ++p) m += s_lds[tid * 65 + p];
    m *= (1.0f / 64.0f);
    float v = 0.f;
    for (int p = 0; p < P; ++p) { float d = s_lds[tid * 65 + p] - m; v += d * d; }
    v *= (1.0f / 64.0f);
    mu_lds[tid] = m;
    rs_lds[tid] = rsqrtf(v + 1e-5f);
  }
  __syncthreads();
  for (int i = 0; i < 4; ++i) {
    int o = tid + 256 * i;
    int t = o >> 6, p = o & 63;
    float sv = s_lds[t * 65 + p];
    s_lds[t * 65 + p] = (sv - mu_lds[t]) * rs_lds[t] * gamma[p] + beta[p];
  }
  __syncthreads();

  // ---------- Phase A4: qk = s (16x64) @ Wqk (64x128); wave w -> 16-col tile w
  {
    v8f acc = {};
    const int n = wave * 16 + lane16;
#pragma unroll
    for (int k0 = 0; k0 < 64; k0 += 4) {
      int ka = k0 + 2 * half;
      v2f a, b;
      a.x = s_lds[lane16 * 65 + ka];
      a.y = s_lds[lane16 * 65 + ka + 1];
      b.x = Wqk[ka * QKD + n];
      b.y = Wqk[(ka + 1) * QKD + n];
      acc = __builtin_amdgcn_wmma_f32_16x16x4_f32(false, a, false, b, (short)0, acc, false, false);
    }
#pragma unroll
    for (int r = 0; r < 8; ++r)
      qk_lds[(r + 8 * half) * 129 + wave * 16 + lane16] = acc[r];
  }
  __syncthreads();

  // ---------- Phase A5: dots = q @ k^T * scale (wave 0; wave-uniform branch)
  if (wave == 0) {
    v8f acc = {};
#pragma unroll
    for (int k0 = 0; k0 < 64; k0 += 4) {
      int ka = k0 + 2 * half;
      v2f a, b;
      a.x = qk_lds[lane16 * 129 + ka];
      a.y = qk_lds[lane16 * 129 + ka + 1];
      b.x = qk_lds[lane16 * 129 + 64 + ka];
      b.y = qk_lds[lane16 * 129 + 64 + ka + 1];
      acc = __builtin_amdgcn_wmma_f32_16x16x4_f32(false, a, false, b, (short)0, acc, false, false);
    }
#pragma unroll
    for (int r = 0; r < 8; ++r)
      attn_lds[(r + 8 * half) * 17 + lane16] = acc[r] * 0.125f;  // DIM_HEAD^-0.5
  }
  __syncthreads();

  // ---------- Phase A6: softmax rows
  if (tid < T) {
    float mx = -3.4e38f;
    for (int j = 0; j < T; ++j) mx = fmaxf(mx, attn_lds[tid * 17 + j]);
    float sum = 0.f;
    for (int j = 0; j < T; ++j) {
      float e = __expf(attn_lds[tid * 17 + j] - mx);
      attn_lds[tid * 17 + j] = e;
      sum += e;
    }
    float inv = 1.0f / sum;
    for (int j = 0; j < T; ++j) attn_lds[tid * 17 + j] *= inv;
  }
  __syncthreads();
  // attn_lds is read-only from here; slabs are per-wave -> no more block barriers.

  // ---------- Phase B: out = attn (16x16) @ v (16x3136) + x, per-wave 64-col chunks
  float* slabp = u.slab + wave * (16 * SLABSTRIDE);

  for (int chunk = wave; chunk < 49; chunk += 8) {   // 49 chunks of 64 columns
    const int n0 = chunk * 64;

#if USE_TDM
    // prior chunk's ds reads must land before the DMA overwrites the slab
    asm volatile("s_wait_dscnt 0x0" ::: "memory");
    {
      unsigned lds_off = (unsigned)(uintptr_t)slabp;               // flat low 32b = LDS byte offset
      unsigned long long ga = (unsigned long long)(uintptr_t)(x + xbase + n0);
      // D# group0: count=1 | lds_addr | global_addr[56:0] | type=2
      u32x4 g0 = { 1u, lds_off, (unsigned)ga,
                   (unsigned)((ga >> 32) & 0x1FFFFFFu) | (2u << 30) };
      // D# group1: data_size=4B, pad_enable, pad_interval=64dw, pad_amount=12dw,
      //            tensor_dim0=HW, tensor_dim1=T, tile=64x16, dim0_stride=CHW
      i32x8 g1 = { (int)((2u << 16) | (1u << 20) | (5u << 22) | (11u << 25)),
                   (int)((unsigned)HW << 16),
                   (int)((unsigned)T << 16) | (int)((unsigned)HW >> 16),
                   (int)(64u << 16),
                   16,
                   (int)(CHW),
                   0, 0 };
      i32x4 gz4 = { 0, 0, 0, 0 };
      i32x8 gz8 = { 0, 0, 0, 0, 0, 0, 0, 0 };
      __builtin_amdgcn_tensor_load_to_lds(g0, g1, gz4, gz4, gz8, 0);
    }
#if __has_builtin(__builtin_amdgcn_s_wait_tensorcnt)
    __builtin_amdgcn_s_wait_tensorcnt(0);
#else
    asm volatile("s_wait_tensorcnt 0x0" ::: "memory");
#endif
    asm volatile("" ::: "memory");   // DMA wrote LDS behind the compiler's back
#else
    // fallback: vectorized staging, 16 rows x 16 float4
#pragma unroll
    for (int i = 0; i < 8; ++i) {
      int idx = lane + 32 * i;
      int row = idx >> 4, c4 = idx & 15;
      *(v4f*)(slabp + row * SLABSTRIDE + c4 * 4) =
          *(const v4f*)(x + xbase + (size_t)row * CHW + n0 + c4 * 4);
    }
#endif

    // 4 sub-tiles of 16 columns: D = attn @ v_sub + x_sub (residual seeds C)
#pragma unroll
    for (int sub = 0; sub < 4; ++sub) {
      const int s16 = sub * 16;
      v8f acc;
#pragma unroll
      for (int r = 0; r < 8; ++r)
        acc[r] = slabp[(r + 8 * half) * SLABSTRIDE + s16 + lane16];
#pragma unroll
      for (int k0 = 0; k0 < 16; k0 += 4) {
        int ka = k0 + 2 * half;
        v2f a, b;
        a.x = attn_lds[lane16 * 17 + ka];
        a.y = attn_lds[lane16 * 17 + ka + 1];
        b.x = slabp[ka * SLABSTRIDE + s16 + lane16];
        b.y = slabp[(ka + 1) * SLABSTRIDE + s16 + lane16];
        acc = __builtin_amdgcn_wmma_f32_16x16x4_f32(false, a, false, b, (short)0, acc, false, false);
      }
      // in-place writeback (reads of this sub-tile precede in program order; LDS is
      // in-order within a wave)
#pragma unroll
      for (int r = 0; r < 8; ++r)
        slabp[(r + 8 * half) * SLABSTRIDE + s16 + lane16] = acc[r];
    }

    // transpose out of LDS and store 128-bit non-temporal (two 256B contiguous rows per instr)
#pragma unroll
    for (int i = 0; i < 8; ++i) {
      int idx = lane + 32 * i;
      int row = idx >> 4, c4 = idx & 15;
      v4f vv = *(const v4f*)(slabp + row * SLABSTRIDE + c4 * 4);
      __builtin_nontemporal_store(vv, (v4f*)(out + xbase + (size_t)row * CHW + n0 + c4 * 4));
    }
  }
}

extern "C" void kernel_launch(void* const* d_in, const int* in_sizes, int n_in,
                              void* d_out, int out_size, void* d_ws, size_t ws_size,
                              hipStream_t stream) {
  (void)in_sizes; (void)n_in; (void)d_ws; (void)ws_size; (void)out_size;
  const float* x     = (const float*)d_in[0];
  const float* pos   = (const float*)d_in[1];
  const float* Wqk   = (const float*)d_in[2];
  const float* gamma = (const float*)d_in[3];
  const float* beta  = (const float*)d_in[4];
  float* out = (float*)d_out;
  cta_attn_kernel<<<dim3(512), dim3(256), 0, stream>>>(x, pos, Wqk, gamma, beta, out);
}